// ProbUCELossHard_CE_15444702397049
// MI455X (gfx1250) — compile-verified
//
#include <hip/hip_runtime.h>

#define N_BINS 15
#define TILE_ROWS 512
#define TILE_FLOATS (TILE_ROWS * 10)   /* 5120 floats  */
#define TILE_BYTES (TILE_FLOATS * 4)   /* 20480 bytes  */
#define THREADS 256
#define CHUNKS_PER_THREAD (TILE_BYTES / (THREADS * 16)) /* 5 x b128 per thread */

/* ---- gfx1250 async-to-LDS path detection ---------------------------------- */
#if defined(__has_builtin)
#if __has_builtin(__builtin_amdgcn_global_load_async_to_lds_b128)
#define HAVE_ASYNC_LDS 1
#endif
#endif
#ifndef HAVE_ASYNC_LDS
#define HAVE_ASYNC_LDS 0
#endif

#if HAVE_ASYNC_LDS
#if defined(__has_builtin) && __has_builtin(__builtin_amdgcn_s_wait_asynccnt)
#define WAIT_ASYNC(n) __builtin_amdgcn_s_wait_asynccnt(n)
#else
#define WAIT_ASYNC(n) asm volatile("s_wait_asynccnt %0" ::"i"(n) : "memory")
#endif
#else
#define WAIT_ASYNC(n) ((void)0)
#endif

typedef int v4i __attribute__((ext_vector_type(4)));
typedef __attribute__((address_space(1))) v4i as1_v4i;  // global
typedef __attribute__((address_space(3))) v4i as3_v4i;  // LDS

// ---------------------------------------------------------------------------
// Kernel 1: zero the 48-float accumulator block in workspace
// ---------------------------------------------------------------------------
__global__ void puce_zero_ws(float* __restrict__ ws) {
  int i = threadIdx.x;
  if (i < 48) ws[i] = 0.0f;
}

// ---------------------------------------------------------------------------
// Kernel 2: tiled streaming histogram with double-buffered LDS staging.
// ---------------------------------------------------------------------------
__global__ __launch_bounds__(THREADS) void puce_hist(const float* __restrict__ logits,
                                                     const int* __restrict__ labels,
                                                     float* __restrict__ ws,
                                                     int n) {
  __shared__ __align__(16) float tiles[2][TILE_FLOATS];
  __shared__ float h[8][3][16];  // per-wave32 bins: count / sum_u / sum_err

  const int tid = threadIdx.x;
  const int wave = tid >> 5;  // wave32
  const float DELTA_F = 1.0f / 15.0f;
  const float LOG2E = 1.4426950408889634f;

  for (int i = tid; i < 8 * 3 * 16; i += THREADS) (&h[0][0][0])[i] = 0.0f;
  __syncthreads();

  // Stage one 20480-byte tile into tiles[buf] (async DMA if available).
  auto issue_tile = [&](long t, int buf) {
    const char* src = (const char*)logits + t * (long)TILE_BYTES;
    char* dst = (char*)&tiles[buf][0];
#if HAVE_ASYNC_LDS
#pragma unroll
    for (int k = 0; k < CHUNKS_PER_THREAD; ++k) {
      int off = (k * THREADS + tid) * 16;
      __builtin_amdgcn_global_load_async_to_lds_b128(
          (as1_v4i*)(uintptr_t)(src + off),
          (as3_v4i*)(uint32_t)(uintptr_t)(dst + off), 0, 0);
    }
#else
#pragma unroll
    for (int k = 0; k < CHUNKS_PER_THREAD; ++k) {
      int off = (k * THREADS + tid) * 16;
      float4 v = *(const float4*)(src + off);         // global_load_b128
      *(float4*)(dst + off) = v;                      // ds_store_b128
    }
#endif
  };

  // Per-row math + binned accumulation (identical numerics to reference).
  auto accum = [&](float2 a0, float2 a1, float2 a2, float2 a3, float2 a4, int lab) {
    float xs[10] = {a0.x, a0.y, a1.x, a1.y, a2.x, a2.y, a3.x, a3.y, a4.x, a4.y};
    float m = xs[0];
    int am = 0;
#pragma unroll
    for (int c = 1; c < 10; ++c)
      if (xs[c] > m) { m = xs[c]; am = c; }
    float s1 = 0.0f, s2 = 0.0f;
#pragma unroll
    for (int c = 0; c < 10; ++c) {
      float e = __builtin_amdgcn_exp2f((xs[c] - m) * LOG2E);
      s1 += e;
      s2 += e * e;
    }
    float p2 = s2 / (s1 * s1);
    float u = -__builtin_amdgcn_logf(p2 + 1e-12f);  // v_log_f32 == log2
    float err = (am != lab) ? 1.0f : 0.0f;
    if (u >= 0.0f && u < 1.0f) {
      int b = (int)(u / DELTA_F);  // floor (u >= 0)
      if (b < N_BINS) {            // drop boundary bin 15 like segment_sum[:15]
        atomicAdd(&h[wave][0][b], 1.0f);  // ds_add_f32
        atomicAdd(&h[wave][1][b], u);
        atomicAdd(&h[wave][2][b], err);
      }
    }
  };

  const long ntiles = (long)(n / TILE_ROWS);
  const long t0 = blockIdx.x;
  int buf = 0;
  if (t0 < ntiles) issue_tile(t0, 0);

  for (long t = t0; t < ntiles; t += gridDim.x) {
    long tn = t + gridDim.x;
    if (tn < ntiles) {
      issue_tile(tn, buf ^ 1);
      WAIT_ASYNC(CHUNKS_PER_THREAD);  // wait for current tile; next stays in flight
    } else {
      WAIT_ASYNC(0);
    }
    long tp = t + 2L * gridDim.x;  // warm L2 two tiles ahead
    if (tp < ntiles)
      __builtin_prefetch((const char*)logits + tp * (long)TILE_BYTES + tid * 80, 0, 0);
    __syncthreads();  // all waves' staged data visible

    const float* base = &tiles[buf][0];
    int2 lb = *(const int2*)(labels + t * TILE_ROWS + 2 * tid);  // coalesced b64
    const float2* r0 = (const float2*)(base + (2 * tid) * 10);
    accum(r0[0], r0[1], r0[2], r0[3], r0[4], lb.x);
    const float2* r1 = (const float2*)(base + (2 * tid + 1) * 10);
    accum(r1[0], r1[1], r1[2], r1[3], r1[4], lb.y);

    __syncthreads();  // done reading buf before it is overwritten
    buf ^= 1;
  }

  // Ragged tail: direct strided loads (rows are 8-byte aligned, 5x b64).
  const long n_full = ntiles * TILE_ROWS;
  const long gstride = (long)gridDim.x * THREADS;
  for (long i = n_full + (long)blockIdx.x * THREADS + tid; i < n; i += gstride) {
    const float2* row = (const float2*)(logits + i * 10);
    accum(row[0], row[1], row[2], row[3], row[4], labels[i]);
  }
  __syncthreads();

  // Merge 8 wave-private histograms -> 48 global float atomics per block.
  if (tid < 48) {
    int which = tid >> 4;
    int bin = tid & 15;
    float acc = 0.0f;
#pragma unroll
    for (int w = 0; w < 8; ++w) acc += h[w][which][bin];
    unsafeAtomicAdd(&ws[which * 16 + bin], acc);  // global_atomic_add_f32
  }
}

// ---------------------------------------------------------------------------
// Kernel 3: 15-bin epilogue -> scalar
// ---------------------------------------------------------------------------
__global__ void puce_final(const float* __restrict__ ws, float* __restrict__ out, int n) {
  if (threadIdx.x == 0 && blockIdx.x == 0) {
    float total = 0.0f;
    float nf = (float)n;
    for (int b = 0; b < N_BINS; ++b) {
      float cnt = ws[b];
      float su = ws[16 + b];
      float se = ws[32 + b];
      bool nonempty = cnt > 0.0f;
      float c = nonempty ? cnt : 1.0f;
      float ub = su / c;
      float eb = se / c;
      float inner = 2.0f * __builtin_amdgcn_exp2f(-ub) - 1.0f;
      inner = (inner < 0.0f) ? 0.0f : inner;
      float r = 0.5f * (1.0f - __builtin_sqrtf(inner));
      float pi = cnt / nf;
      total += nonempty ? __builtin_fabsf(eb - r) * pi : 0.0f;
    }
    out[0] = total;
  }
}

// ---------------------------------------------------------------------------
// Host-side launcher (graph-capture safe: only kernel launches on `stream`)
// ---------------------------------------------------------------------------
extern "C" void kernel_launch(void* const* d_in, const int* in_sizes, int n_in,
                              void* d_out, int out_size, void* d_ws, size_t ws_size,
                              hipStream_t stream) {
  const float* logits = (const float*)d_in[0];
  const int* labels = (const int*)d_in[1];
  const int n = in_sizes[1];
  float* ws = (float*)d_ws;

  puce_zero_ws<<<1, 64, 0, stream>>>(ws);

  long ntiles = (long)n / TILE_ROWS;
  int blocks = (int)(ntiles < 1 ? 1 : (ntiles > 2048 ? 2048 : ntiles));
  puce_hist<<<blocks, THREADS, 0, stream>>>(logits, labels, ws, n);

  puce_final<<<1, 32, 0, stream>>>(ws, (float*)d_out, n);
}